// F_B_2D_80135499809047
// MI455X (gfx1250) — compile-verified
//
#include <hip/hip_runtime.h>

// ---------------------------------------------------------------------------
// DimeNet spherical basis: out[t, l*6+n] = env(d[e]) * NORM[l,n] *
//   j_l(Z[l,n]*d[e]/5) * Y_NORM[l] * P_l(cos(Angles[t])),  e = edge_index[t]
// Memory-bound: 672MB output.  Strategy:
//   k0: compute Bessel zeros Z + NORM on device (once per launch, ~us)
//   k1: rbf_env table (E x 42) in d_ws  (168MB -> L2-resident on MI455X)
//   k2: per 256-t tile: cbf via V_WMMA_F32_16X16X4_F32, coalesced float2 out
// Cache policy: output stores + Angles/idx loads are non-temporal (TH=NT) so
// the 672MB output stream does not evict the L2-resident rbf table.
// ---------------------------------------------------------------------------

#if defined(__HIP_DEVICE_COMPILE__) && __has_builtin(__builtin_amdgcn_wmma_f32_16x16x4_f32)
#define HAVE_WMMA 1
#else
#define HAVE_WMMA 0
#endif

typedef __attribute__((ext_vector_type(2))) float v2f;
typedef __attribute__((ext_vector_type(8))) float v8f;

// Y_NORM[l] = sqrt((2l+1)/(4*pi))
#define Y0 0.2820947917738781f
#define Y1 0.4886025119029199f
#define Y2 0.6307831305050401f
#define Y3 0.7463526651802308f
#define Y4 0.8462843753216344f
#define Y5 0.9356025796273889f
#define Y6 1.0171072362820548f

// W[p][l] = Y_NORM[l] * (coefficient of c^p in Legendre P_l), padded to N=16
__device__ __constant__ float WPAD[8][16] = {
  { Y0,  0.f, -0.5f*Y2,  0.f,      0.375f*Y4,  0.f,       -0.3125f*Y6,  0,0,0,0,0,0,0,0,0 },
  { 0.f, Y1,  0.f,      -1.5f*Y3,  0.f,        1.875f*Y5,  0.f,         0,0,0,0,0,0,0,0,0 },
  { 0.f, 0.f, 1.5f*Y2,   0.f,     -3.75f*Y4,   0.f,        6.5625f*Y6,  0,0,0,0,0,0,0,0,0 },
  { 0.f, 0.f, 0.f,       2.5f*Y3,  0.f,       -8.75f*Y5,   0.f,         0,0,0,0,0,0,0,0,0 },
  { 0.f, 0.f, 0.f,       0.f,      4.375f*Y4,  0.f,      -19.6875f*Y6,  0,0,0,0,0,0,0,0,0 },
  { 0.f, 0.f, 0.f,       0.f,      0.f,        7.875f*Y5,  0.f,         0,0,0,0,0,0,0,0,0 },
  { 0.f, 0.f, 0.f,       0.f,      0.f,        0.f,       14.4375f*Y6,  0,0,0,0,0,0,0,0,0 },
  { 0.f, 0.f, 0.f,       0.f,      0.f,        0.f,        0.f,         0,0,0,0,0,0,0,0,0 },
};

__device__ __forceinline__ float fast_rcp(float x) {
#if defined(__HIP_DEVICE_COMPILE__)
  return __builtin_amdgcn_rcpf(x);
#else
  return 1.0f / x;
#endif
}

// float spherical Bessel j_l via upward recurrence (bisection helper)
__device__ __forceinline__ float jn_f(int l, float x) {
  float s, c;
  __sincosf(x, &s, &c);
  float rx = fast_rcp(x);
  float j0 = s * rx;
  if (l == 0) return j0;
  float j1 = (j0 - c) * rx;
  for (int i = 2; i <= l; ++i) {
    float jn = (float)(2 * i - 1) * rx * j1 - j0;
    j0 = j1; j1 = jn;
  }
  return j1;
}

// double j_{l-1}, j_l pair (Newton polish + NORM); j_{-1}(x) = cos(x)/x
__device__ void jn_pair_d(int l, double x, double* jlm1, double* jl) {
  double s = sin(x), c = cos(x);
  double j0 = s / x;
  if (l == 0) { *jlm1 = c / x; *jl = j0; return; }
  double j1 = s / (x * x) - c / x;
  for (int i = 2; i <= l; ++i) {
    double jn = (double)(2 * i - 1) / x * j1 - j0;
    j0 = j1; j1 = jn;
  }
  *jlm1 = j0; *jl = j1;
}

// ---- Kernel 0: Z (7x6) and NORM (7x6), one 64-thread block -----------------
__global__ __launch_bounds__(64) void init_zn_kernel(float* __restrict__ Zc,
                                                     float* __restrict__ Nc) {
  __shared__ float prev[13];
  __shared__ float Zs[7][6];
  const int tid = threadIdx.x;
  const float PI_F = 3.14159265358979323846f;
  if (tid < 12) prev[tid] = (float)(tid + 1) * PI_F;
  __syncthreads();
  if (tid < 6) Zs[0][tid] = prev[tid];
  int len = 12;
  for (int l = 1; l < 7; ++l) {
    float z = 0.f;
    const bool active = tid < (len - 1);
    if (active) {
      float a = prev[tid], b = prev[tid + 1];
      float fa = jn_f(l, a);
      for (int it = 0; it < 40; ++it) {
        float m = 0.5f * (a + b);
        float fm = jn_f(l, m);
        if ((fm >= 0.f) == (fa >= 0.f)) { a = m; fa = fm; } else { b = m; }
      }
      z = 0.5f * (a + b);
    }
    __syncthreads();
    if (active) prev[tid] = z;
    __syncthreads();
    if (tid < 6) Zs[l][tid] = prev[tid];
    len -= 1;
    __syncthreads();
  }
  if (tid < 42) {
    const int l = tid / 6, n2 = tid % 6;
    double z = (double)Zs[l][n2];
    for (int it = 0; it < 3; ++it) {           // Newton: j_l' = j_{l-1} - (l+1)/x j_l
      double jlm1, jl;
      jn_pair_d(l, z, &jlm1, &jl);
      z -= jl / (jlm1 - (double)(l + 1) / z * jl);
    }
    Zc[tid] = (float)z;
    double jl2, jlp1;
    jn_pair_d(l + 1, z, &jl2, &jlp1);
    Nc[tid] = (float)(1.4142135623730951 / fabs(jlp1));
  }
}

// ---- rbf_env row (42 values) for one distance ------------------------------
__device__ __forceinline__ void rbf_row(float dv, const float* __restrict__ Zc,
                                        const float* __restrict__ Nc,
                                        float* row) {
  const float u = dv * 0.2f;                       // d / CUTOFF (== d / ENV_CUTOFF)
  const float u2 = u * u;
  const float u5 = u2 * u2 * u;
  float env = fast_rcp(u) + u5 * (-28.0f + u * (48.0f - 21.0f * u));
  if (u >= 1.0f) env = 0.0f;
  #pragma unroll
  for (int l = 0; l < 7; ++l) {
    #pragma unroll
    for (int n2 = 0; n2 < 6; ++n2) {
      const float a = Zc[l * 6 + n2] * u;
      float s, c;
      __sincosf(a, &s, &c);
      const float ra = fast_rcp(a);
      float j0 = s * ra;
      float j;
      if (l == 0) {
        j = j0;
      } else {
        float j1 = (j0 - c) * ra;
        #pragma unroll
        for (int i = 2; i <= l; ++i) {
          float jn = (float)(2 * i - 1) * ra * j1 - j0;
          j0 = j1; j1 = jn;
        }
        j = j1;
      }
      row[l * 6 + n2] = env * Nc[l * 6 + n2] * j;
    }
  }
}

// ---- Kernel 1: build rbf_env table (E x 42), regular-temporal (L2) ---------
__global__ __launch_bounds__(256) void build_rbf_kernel(
    const float* __restrict__ d, const float* __restrict__ Zc,
    const float* __restrict__ Nc, v2f* __restrict__ table, int E_) {
  const int e = blockIdx.x * 256 + threadIdx.x;
  if (e >= E_) return;
  float row[42];
  rbf_row(d[e], Zc, Nc, row);
  v2f* dst = table + (size_t)e * 21;
  #pragma unroll
  for (int k = 0; k < 21; ++k) {
    v2f w; w.x = row[2 * k]; w.y = row[2 * k + 1];
    dst[k] = w;
  }
}

// ---- Kernel 2: gather + WMMA Legendre + coalesced NT combine ---------------
// LDS: c_s[256] @0, idx_s[256] @1024, cbf_s[256*9] @2048, rbf_s[256*42] @11264
template <bool USE_TABLE>
__global__ __launch_bounds__(256) void combine_kernel(
    const float* __restrict__ Angles, const int* __restrict__ eidx,
    const v2f* __restrict__ table, const float* __restrict__ darr,
    const float* __restrict__ Zc, const float* __restrict__ Nc,
    v2f* __restrict__ out, int T_) {
  extern __shared__ char smem_raw[];
  float* c_s   = (float*)smem_raw;
  int*   idx_s = (int*)(smem_raw + 1024);
  float* cbf_s = (float*)(smem_raw + 2048);
  float* rbf_s = (float*)(smem_raw + 2048 + 9216);

  const int tid = threadIdx.x;
  const int t = blockIdx.x * 256 + tid;
  const int tc = (t < T_) ? t : (T_ - 1);          // clamp (no early return: EXEC stays full)
  const float cv = __cosf(__builtin_nontemporal_load(&Angles[tc]));
  c_s[tid] = cv;
  const int e = __builtin_nontemporal_load(&eidx[tc]);
  idx_s[tid] = e;
  if (!USE_TABLE) rbf_row(darr[e], Zc, Nc, rbf_s + tid * 42);
  __syncthreads();

#if HAVE_WMMA
  {
    // cbf[16 t's][7 l's] = Cpow(16x8) @ WPAD(8x7) via two K=4 f32 WMMAs
    const int lane = tid & 31;
    const int n = lane & 15;                       // N (= l) index
    const bool hi = lane >= 16;
    const int waveBase = (tid >> 5) * 32;
    #pragma unroll
    for (int g = 0; g < 2; ++g) {
      const int baseT = waveBase + g * 16;
      const float c = c_s[baseT + n];              // row M = n
      const float c2 = c * c, c3 = c2 * c, c4 = c2 * c2, c5 = c4 * c, c6 = c4 * c2;
      v2f A, A2, B, B2;
      // A 16x4 f32 layout: VGPR0 = K0 (lanes 0-15) / K2 (16-31); VGPR1 = K1/K3
      A.x  = hi ? c2 : 1.0f;  A.y  = hi ? c3 : c;
      A2.x = hi ? c6 : c4;    A2.y = hi ? 0.0f : c5;   // p=7 row is zero
      B.x  = hi ? WPAD[2][n] : WPAD[0][n];
      B.y  = hi ? WPAD[3][n] : WPAD[1][n];
      B2.x = hi ? WPAD[6][n] : WPAD[4][n];
      B2.y = hi ? 0.0f       : WPAD[5][n];
      v8f acc = {0.f, 0.f, 0.f, 0.f, 0.f, 0.f, 0.f, 0.f};
      acc = __builtin_amdgcn_wmma_f32_16x16x4_f32(false, A,  false, B,  (short)0, acc, false, false);
      acc = __builtin_amdgcn_wmma_f32_16x16x4_f32(false, A2, false, B2, (short)0, acc, false, false);
      if (n < 7) {
        #pragma unroll
        for (int v = 0; v < 8; ++v) {              // C layout: VGPR v -> M=v (lo) / M=v+8 (hi)
          const int m = v + (hi ? 8 : 0);
          cbf_s[(baseT + m) * 9 + n] = acc[v];
        }
      }
    }
  }
#else
  {
    const float YNs[7] = {Y0, Y1, Y2, Y3, Y4, Y5, Y6};
    float pm2 = 1.f, pm1 = cv;
    cbf_s[tid * 9 + 0] = Y0;
    cbf_s[tid * 9 + 1] = Y1 * cv;
    #pragma unroll
    for (int l = 2; l < 7; ++l) {
      const float p = ((float)(2 * l - 1) * cv * pm1 - (float)(l - 1) * pm2) * (1.0f / (float)l);
      cbf_s[tid * 9 + l] = YNs[l] * p;
      pm2 = pm1; pm1 = p;
    }
  }
#endif
  __syncthreads();

  // Flat, coalesced output phase: 256 rows * 21 float2 each.
  // Output stores are non-temporal so the streaming 672MB does not evict the
  // L2-resident rbf table (168MB vs 192MB L2).
  const int rowBase = blockIdx.x * 256;
  for (int j = tid; j < 256 * 21; j += 256) {
    const int tl = j / 21;
    const int k2 = j - tl * 21;
    const int l = k2 / 3;                          // (2*k2)/6
    const float s = cbf_s[tl * 9 + l];
    v2f r;
    if (USE_TABLE) {
      const size_t ee = (size_t)(unsigned)idx_s[tl];
      r = table[ee * 21 + k2];                     // regular-temporal: keep in L2
    } else {
      r = ((const v2f*)rbf_s)[tl * 21 + k2];
    }
    const int tg = rowBase + tl;
    v2f o; o.x = r.x * s; o.y = r.y * s;
    if (tg < T_) __builtin_nontemporal_store(o, out + (size_t)tg * 21 + k2);
  }
}

// ---------------------------------------------------------------------------
extern "C" void kernel_launch(void* const* d_in, const int* in_sizes, int n_in,
                              void* d_out, int out_size, void* d_ws, size_t ws_size,
                              hipStream_t stream) {
  (void)n_in; (void)out_size;
  const float* d      = (const float*)d_in[0];
  const float* Angles = (const float*)d_in[1];
  const int*   eidx   = (const int*)d_in[2];
  const int E_ = in_sizes[0];
  const int T_ = in_sizes[1];

  float* Zc    = (float*)d_ws;        // 42 floats
  float* Nc    = Zc + 48;             // 42 floats
  float* table = Zc + 96;             // E*42 floats (16B aligned: 384B offset)
  const size_t need = 96 * sizeof(float) + (size_t)E_ * 42 * sizeof(float);
  const bool use_table = ws_size >= need;

  init_zn_kernel<<<1, 64, 0, stream>>>(Zc, Nc);

  const int tb = (T_ + 255) / 256;
  if (use_table) {
    build_rbf_kernel<<<(E_ + 255) / 256, 256, 0, stream>>>(d, Zc, Nc, (v2f*)table, E_);
    combine_kernel<true><<<tb, 256, 2048 + 9216, stream>>>(
        Angles, eidx, (const v2f*)table, d, Zc, Nc, (v2f*)d_out, T_);
  } else {
    combine_kernel<false><<<tb, 256, 2048 + 9216 + 256 * 42 * 4, stream>>>(
        Angles, eidx, nullptr, d, Zc, Nc, (v2f*)d_out, T_);
  }
}